// PNA_81973745812096
// MI455X (gfx1250) — compile-verified
//
#include <hip/hip_runtime.h>

// ---------------------------------------------------------------------------
// PNA conv, fused, fp32 WMMA (V_WMMA_F32_16X16X4_F32), gfx1250 wave32.
//   m = x[dst]@W1 + x[src]@W2 + ea@(W_e@W3) + bcomb          (edge kernel)
//   scatter: s1+=m, s2+=m*m, mx=max, mn=min (ordered-uint), deg++
//   out = [x | aggr | sc1*aggr | sc2*aggr] @ (W_post@W_lin) + bout
//         (node kernel; scalers folded into the WMMA A fragments)
// Weights are stored pair-interleaved (rows r,r+1 adjacent per column) so a
// WMMA B fragment is one aligned b64 load. Edge-kernel weights are staged
// global->LDS by the Tensor Data Mover (tensor_load_to_lds + TENSORcnt).
// ---------------------------------------------------------------------------

#define NN      50000
#define NE      800000
#define FD      64
#define NF      (NN * FD)
#define EPS     1e-5f
#define NTILE_E (NE / 16)     // 50000
#define NTILE_N (NN / 16)     // 3125
#define EDGEW_N (3 * FD * FD + FD)  // 12352 floats: W1|W2|We3|bcomb (swizzled)

typedef __attribute__((ext_vector_type(2))) float    v2f;
typedef __attribute__((ext_vector_type(8))) float    v8f;
typedef __attribute__((ext_vector_type(4))) unsigned u32x4;
typedef __attribute__((ext_vector_type(4))) int      i32x4;
typedef __attribute__((ext_vector_type(8))) int      i32x8;

__device__ inline v8f wmma_f32(v2f a, v2f b, v8f c) {
  // D = A(16x4,f32) x B(4x16,f32) + C(16x16,f32); fp32 FMA, RNE — matches ref.
  return __builtin_amdgcn_wmma_f32_16x16x4_f32(
      /*neg_a=*/false, a, /*neg_b=*/false, b,
      /*c_mod=*/(short)0, c, /*reuse_a=*/false, /*reuse_b=*/false);
}

// Order-preserving float<->uint map: monotone, so uint atomicMax/Min == float
// max/min, and (unlike fp-add atomics) fully deterministic.
__device__ inline unsigned f2ord(float f) {
  unsigned u = __float_as_uint(f);
  return (u & 0x80000000u) ? ~u : (u | 0x80000000u);
}
__device__ inline float ord2f(unsigned u) {
  return (u & 0x80000000u) ? __uint_as_float(u & 0x7fffffffu)
                           : __uint_as_float(~u);
}

// pair-interleaved ("swizzled") index: rows r (even) and r+1 adjacent per col
__device__ __host__ inline int swz(int r, int c) {
  return (r >> 1) * 128 + c * 2 + (r & 1);
}

// ---------------------------------------------------------------------------
// workspace init: s1=s2=0, mx=ord(-inf)=0, mn=ord(+inf)=~0, deg=0
__global__ void k_init(float* s1, float* s2, unsigned* mx, unsigned* mn,
                       unsigned* deg) {
  int i = blockIdx.x * blockDim.x + threadIdx.x;
  if (i < NF) { s1[i] = 0.f; s2[i] = 0.f; mx[i] = 0u; mn[i] = 0xFFFFFFFFu; }
  if (i < NN) deg[i] = 0u;
}

// Pack edgeW = [W1 | W2 | W_e@W3 | bcomb], all matrices pair-interleaved.
__global__ void k_fuse_edge(const float* __restrict__ We,
                            const float* __restrict__ be,
                            const float* __restrict__ Wpre,
                            const float* __restrict__ bpre,
                            float* __restrict__ edgeW) {
  int t = blockIdx.x * blockDim.x + threadIdx.x;
  if (t < 2 * FD * FD) {        // W1 (Wpre rows 0..63), W2 (rows 64..127)
    int r = t >> 6, c = t & 63;
    int m = r >> 6, rr = r & 63;
    edgeW[m * FD * FD + swz(rr, c)] = Wpre[t];
  }
  if (t < FD * FD) {            // We3 = W_e @ W_pre[128:192]
    int r = t >> 6, c = t & 63;
    float s = 0.f;
    for (int k = 0; k < FD; ++k) s += We[r * FD + k] * Wpre[(128 + k) * FD + c];
    edgeW[2 * FD * FD + swz(r, c)] = s;
  }
  if (t < FD) {                 // bcomb = b_pre + b_e @ W_pre[128:192]
    float s = bpre[t];
    for (int k = 0; k < FD; ++k) s += be[k] * Wpre[(128 + k) * FD + t];
    edgeW[3 * FD * FD + t] = s;
  }
}

// Wcs = swizzle(W_post @ W_lin) (832x64), bout = b_lin + b_post @ W_lin
__global__ void k_fuse_post(const float* __restrict__ Wpost,
                            const float* __restrict__ bpost,
                            const float* __restrict__ Wlin,
                            const float* __restrict__ blin,
                            float* __restrict__ Wcs, float* __restrict__ bo) {
  int t = blockIdx.x * blockDim.x + threadIdx.x;
  if (t < 13 * FD * FD) {
    int r = t >> 6, c = t & 63;
    float s = 0.f;
    for (int k = 0; k < FD; ++k) s += Wpost[r * FD + k] * Wlin[k * FD + c];
    Wcs[swz(r, c)] = s;
  }
  if (t < FD) {
    float s = blin[t];
    for (int k = 0; k < FD; ++k) s += bpost[k] * Wlin[k * FD + t];
    bo[t] = s;
  }
}

// ---------------------------------------------------------------------------
// Edge kernel: 8 waves/block, one 16-edge tile per wave.
// edgeW (48.25 KB) staged into LDS once per block by the TDM.
__global__ __launch_bounds__(256) void k_edge(
    const float* __restrict__ x, const long long* __restrict__ ei,
    const float* __restrict__ ea, const float* __restrict__ edgeW,
    float* __restrict__ s1, float* __restrict__ s2,
    unsigned* __restrict__ mx, unsigned* __restrict__ mn,
    unsigned* __restrict__ deg) {
  __shared__ float lw[EDGEW_N];
  __shared__ unsigned sdst[8 * 16];
  __shared__ unsigned ssrc[8 * 16];

  const int tid = threadIdx.x;
  const int wave = tid >> 5, lane = tid & 31;
  const int ll = lane & 15, hl = lane >> 4;
  const int tile = blockIdx.x * 8 + wave;

  if (wave == 0) {
    // TDM: 1 x EDGEW_N row of 4-byte elements, global -> LDS (one issue/wave).
    const unsigned ldsoff = (unsigned)(size_t)(void*)lw;
    const unsigned long long ga = (unsigned long long)(size_t)edgeW;
    u32x4 g0 = {1u,                                    // count=1 descriptor
                ldsoff,                                // lds_addr (bytes)
                (unsigned)ga,                          // global_addr[31:0]
                (unsigned)((ga >> 32) & 0x01FFFFFFull) // global_addr[56:32]
                    | 0x80000000u};                    // type=2 ("image")
    i32x8 g1 = {0x00020000,                            // data_size=4B
                (int)(((unsigned)EDGEW_N & 0xFFFFu) << 16),  // tensor_dim0 lo
                0x00010000,                            // tensor_dim1 = 1
                (int)(((unsigned)EDGEW_N & 0xFFFFu) << 16),  // tile_dim0
                1,                                     // tile_dim1 = 1
                EDGEW_N,                               // tensor_dim0_stride
                0, 0};
    i32x4 z4 = {0, 0, 0, 0};
#if defined(__clang_major__) && __clang_major__ >= 23
    i32x8 z8 = {0, 0, 0, 0, 0, 0, 0, 0};
    __builtin_amdgcn_tensor_load_to_lds(g0, g1, z4, z4, z8, 0);
#else
    __builtin_amdgcn_tensor_load_to_lds(g0, g1, z4, z4, 0);
#endif
    __builtin_amdgcn_s_wait_tensorcnt(0);
  }
  if (tile < NTILE_E && lane < 16) {
    const int e = tile * 16 + lane;
    ssrc[wave * 16 + lane] = (unsigned)ei[e];        // row 0: src (x_j)
    sdst[wave * 16 + lane] = (unsigned)ei[NE + e];   // row 1: dst (x_i)
  }
  __syncthreads();
  if (tile >= NTILE_E) return;

  v8f acc[4];
#pragma unroll
  for (int nt = 0; nt < 4; ++nt) {
    const float b = lw[3 * FD * FD + nt * 16 + ll];
    v8f t = {b, b, b, b, b, b, b, b};
    acc[nt] = t;
  }

  // A fragments: lane holds row M=ll, K-cols {4k+2*hl, +1}  (ISA 7.12.2)
  const float* aD = x + (size_t)sdst[wave * 16 + ll] * FD + 2 * hl;
  const float* aS = x + (size_t)ssrc[wave * 16 + ll] * FD + 2 * hl;
  const float* aE = ea + (size_t)(tile * 16 + ll) * FD + 2 * hl;

  for (int k = 0; k < 16; ++k) {
    v2f ad = *(const v2f*)(aD + 4 * k);
    v2f as = *(const v2f*)(aS + 4 * k);
    v2f ae = *(const v2f*)(aE + 4 * k);
    const float* bb = lw + (2 * k + hl) * 128;  // pair-row (4k+2hl)>>1
#pragma unroll
    for (int nt = 0; nt < 4; ++nt) {
      const int co = (nt * 16 + ll) * 2;
      v2f b1 = *(const v2f*)(bb + co);                 // W1
      v2f b2 = *(const v2f*)(bb + FD * FD + co);       // W2
      v2f b3 = *(const v2f*)(bb + 2 * FD * FD + co);   // We3
      acc[nt] = wmma_f32(ad, b1, acc[nt]);
      acc[nt] = wmma_f32(as, b2, acc[nt]);
      acc[nt] = wmma_f32(ae, b3, acc[nt]);
    }
  }

  // 4-way scatter-reduce. C/D layout: VGPR v, lane -> (M=v+8*hl, N=nt*16+ll)
#pragma unroll
  for (int nt = 0; nt < 4; ++nt) {
    const int col = nt * 16 + ll;
#pragma unroll
    for (int v = 0; v < 8; ++v) {
      const float val = acc[nt][v];
      const unsigned node = sdst[wave * 16 + v + 8 * hl];
      const size_t off = (size_t)node * FD + col;
      atomicAdd(&s1[off], val);
      atomicAdd(&s2[off], val * val);
      atomicMax(&mx[off], f2ord(val));
      atomicMin(&mn[off], f2ord(val));
    }
  }
  if (lane < 16) atomicAdd(&deg[sdst[wave * 16 + lane]], 1u);
}

// ---------------------------------------------------------------------------
// Deterministic mean of log(deg+1): one block, LDS tree reduction.
__global__ __launch_bounds__(1024) void k_avglog(const unsigned* __restrict__ deg,
                                                 float* __restrict__ avg) {
  __shared__ float red[1024];
  float s = 0.f;
  for (int i = threadIdx.x; i < NN; i += 1024) s += logf((float)deg[i] + 1.0f);
  red[threadIdx.x] = s;
  __syncthreads();
  for (int w = 512; w > 0; w >>= 1) {
    if ((int)threadIdx.x < w) red[threadIdx.x] += red[threadIdx.x + w];
    __syncthreads();
  }
  if (threadIdx.x == 0) avg[0] = red[0] / (float)NN;
}

// ---------------------------------------------------------------------------
// Node kernel: one 16-node tile per wave, 16x832 @ 832x64 via fp32 WMMA.
// mean/std/max/min A fragments synthesized from s1/s2/mx/mn/deg; degree
// scalers are per-row == per-lane, folded into scaled A fragments.
__global__ __launch_bounds__(256) void k_node(
    const float* __restrict__ x, const float* __restrict__ s1,
    const float* __restrict__ s2, const unsigned* __restrict__ mx,
    const unsigned* __restrict__ mn, const unsigned* __restrict__ deg,
    const float* __restrict__ Wcs, const float* __restrict__ bout,
    const float* __restrict__ avgp, float* __restrict__ out) {
  const int tid = threadIdx.x;
  const int wave = tid >> 5, lane = tid & 31;
  const int ll = lane & 15, hl = lane >> 4;
  const int tile = blockIdx.x * 8 + wave;
  if (tile >= NTILE_N) return;
  const int n0 = tile * 16;
  const int nodeA = n0 + ll;  // A-fragment row for this lane

  const float avg = avgp[0];
  const unsigned d = deg[nodeA];
  const bool hasb = d > 0u;
  const float degc = fmaxf((float)d, 1.0f);
  const float inv = 1.0f / degc;
  const float ld = logf(degc + 1.0f);
  const float sc1 = ld / avg;   // amplification
  const float sc2 = avg / ld;   // attenuation

  v8f acc[4];
#pragma unroll
  for (int nt = 0; nt < 4; ++nt) {
    const float b = bout[nt * 16 + ll];
    v8f t = {b, b, b, b, b, b, b, b};
    acc[nt] = t;
  }

  // Block 0: x (Wcs pair-rows 0..31)
  const float* ax = x + (size_t)nodeA * FD + 2 * hl;
  for (int k = 0; k < 16; ++k) {
    v2f a = *(const v2f*)(ax + 4 * k);
    const float* bb = Wcs + (2 * k + hl) * 128;
#pragma unroll
    for (int nt = 0; nt < 4; ++nt) {
      v2f b = *(const v2f*)(bb + (nt * 16 + ll) * 2);
      acc[nt] = wmma_f32(a, b, acc[nt]);
    }
  }

  // Blocks 1..3: aggr (rows 64..319), sc1*aggr (320..575), sc2*aggr (576..831)
  const size_t rb = (size_t)nodeA * FD;
  for (int k = 0; k < 64; ++k) {
    const int c0 = 4 * k + 2 * hl;  // aggr column, 0..255 (even)
    const int seg = k >> 4;         // 0=mean 1=std 2=max 3=min (wave-uniform)
    const int f = c0 & 63;
    v2f a;
    if (seg == 0) {
      v2f t = *(const v2f*)(s1 + rb + f);
      a = t * inv;
    } else if (seg == 1) {
      v2f t1 = *(const v2f*)(s1 + rb + f);
      v2f t2 = *(const v2f*)(s2 + rb + f);
      v2f mean = t1 * inv;
      v2f var = t2 * inv - mean * mean;
      a.x = sqrtf(fmaxf(var.x, 0.f) + EPS);
      a.y = sqrtf(fmaxf(var.y, 0.f) + EPS);
    } else if (seg == 2) {
      a.x = hasb ? ord2f(mx[rb + f]) : 0.f;
      a.y = hasb ? ord2f(mx[rb + f + 1]) : 0.f;
    } else {
      a.x = hasb ? ord2f(mn[rb + f]) : 0.f;
      a.y = hasb ? ord2f(mn[rb + f + 1]) : 0.f;
    }
    v2f a1 = a * sc1;
    v2f a2 = a * sc2;
    const int pr = 2 * k + hl;      // c0>>1
    const float* b1p = Wcs + (32 + pr) * 128;    // rows  64+c0
    const float* b2p = Wcs + (160 + pr) * 128;   // rows 320+c0
    const float* b3p = Wcs + (288 + pr) * 128;   // rows 576+c0
#pragma unroll
    for (int nt = 0; nt < 4; ++nt) {
      const int co = (nt * 16 + ll) * 2;
      v2f b1 = *(const v2f*)(b1p + co);
      v2f b2 = *(const v2f*)(b2p + co);
      v2f b3 = *(const v2f*)(b3p + co);
      acc[nt] = wmma_f32(a, b1, acc[nt]);
      acc[nt] = wmma_f32(a1, b2, acc[nt]);
      acc[nt] = wmma_f32(a2, b3, acc[nt]);
    }
  }

#pragma unroll
  for (int nt = 0; nt < 4; ++nt) {
#pragma unroll
    for (int v = 0; v < 8; ++v) {
      out[(size_t)(n0 + v + 8 * hl) * FD + nt * 16 + ll] = acc[nt][v];
    }
  }
}

// ---------------------------------------------------------------------------
extern "C" void kernel_launch(void* const* d_in, const int* in_sizes, int n_in,
                              void* d_out, int out_size, void* d_ws,
                              size_t ws_size, hipStream_t stream) {
  const float* x      = (const float*)d_in[0];
  const long long* ei = (const long long*)d_in[1];  // int64 [2, E]
  const float* ea     = (const float*)d_in[2];
  const float* We     = (const float*)d_in[3];
  const float* be     = (const float*)d_in[4];
  const float* Wpre   = (const float*)d_in[5];
  const float* bpre   = (const float*)d_in[6];
  const float* Wpost  = (const float*)d_in[7];
  const float* bpost  = (const float*)d_in[8];
  const float* Wlin   = (const float*)d_in[9];
  const float* blin   = (const float*)d_in[10];
  float* out = (float*)d_out;

  // workspace layout (re-initialized every call; ~51.6 MB)
  char* w = (char*)d_ws;
  float*    s1    = (float*)w;                w += (size_t)NF * 4;
  float*    s2    = (float*)w;                w += (size_t)NF * 4;
  unsigned* mx    = (unsigned*)w;             w += (size_t)NF * 4;
  unsigned* mn    = (unsigned*)w;             w += (size_t)NF * 4;
  unsigned* deg   = (unsigned*)w;             w += (size_t)NN * 4;
  float*    edgeW = (float*)w;                w += (size_t)EDGEW_N * 4;
  float*    Wcs   = (float*)w;                w += (size_t)13 * FD * FD * 4;
  float*    bo    = (float*)w;                w += (size_t)FD * 4;
  float*    avg   = (float*)w;

  k_init<<<(NF + 255) / 256, 256, 0, stream>>>(s1, s2, mx, mn, deg);
  k_fuse_edge<<<(2 * FD * FD + 255) / 256, 256, 0, stream>>>(We, be, Wpre,
                                                             bpre, edgeW);
  k_fuse_post<<<(13 * FD * FD + 255) / 256, 256, 0, stream>>>(Wpost, bpost,
                                                              Wlin, blin, Wcs,
                                                              bo);
  k_edge<<<NTILE_E / 8, 256, 0, stream>>>(x, ei, ea, edgeW, s1, s2, mx, mn,
                                          deg);
  k_avglog<<<1, 1024, 0, stream>>>(deg, avg);
  k_node<<<(NTILE_N + 7) / 8, 256, 0, stream>>>(x, s1, s2, mx, mn, deg, Wcs,
                                                bo, avg, out);
}